// GATModel_30975304139311
// MI455X (gfx1250) — compile-verified
//
#include <hip/hip_runtime.h>
#include <math.h>

typedef __attribute__((ext_vector_type(2))) float v2f;
typedef __attribute__((ext_vector_type(8))) float v8f;

#define NEG_SLOPE 0.2f

// ---------------------------------------------------------------------------
// GEMM: h[n, j] = sum_k x[n,k] * W[j,k]; x:[N,128], W:[128,128] row-major.
// One block = 256 threads = 8 waves; block computes 16 rows x 128 cols.
// Wave w computes the 16x16 tile at columns [16w, 16w+16) using
// v_wmma_f32_16x16x4_f32 chained over K=128 (32 WMMAs).
// ---------------------------------------------------------------------------
__global__ __launch_bounds__(256) void k_gemm_xWT(const float* __restrict__ x,
                                                  const float* __restrict__ W,
                                                  float* __restrict__ h)
{
    __shared__ float xs[16 * 128];
    const int tid  = threadIdx.x;
    const int lane = tid & 31;
    const int wave = tid >> 5;
    const long rowBase = (long)blockIdx.x * 16;

    for (int i = tid; i < 16 * 128; i += 256) {
        int r = i >> 7, c = i & 127;
        xs[i] = x[(rowBase + r) * 128 + c];
    }
    __syncthreads();

    // A 16x4 f32 layout: lanes 0-15 -> M=lane, VGPR0=K0,VGPR1=K1;
    //                    lanes 16-31 -> M=lane-16, VGPR0=K2,VGPR1=K3.
    // B 4x16 mirrors this with N instead of M.
    const int m16     = lane & 15;
    const int khalf   = (lane >> 4) << 1;     // 0 or 2
    const int colBase = wave * 16;

    v8f acc = {};
    #pragma unroll
    for (int k0 = 0; k0 < 128; k0 += 4) {
        v2f a, b;
        a.x = xs[m16 * 128 + k0 + khalf];
        a.y = xs[m16 * 128 + k0 + khalf + 1];
        const float* wr = W + (long)(colBase + m16) * 128 + k0 + khalf;
        b.x = wr[0];
        b.y = wr[1];
        acc = __builtin_amdgcn_wmma_f32_16x16x4_f32(false, a, false, b,
                                                    (short)0, acc, false, false);
    }
    // D 16x16 f32: VGPR r -> M = r + 8*(lane>=16), N = lane&15
    const long rOff = rowBase + ((lane >> 4) << 3);
    const int  col  = colBase + m16;
    #pragma unroll
    for (int r = 0; r < 8; ++r)
        h[(rOff + r) * 128 + col] = acc[r];
}

// ---------------------------------------------------------------------------
// Per (node, head) attention coefficients: as = <h[n,hd,:], a_src[hd,:]> etc.
// ---------------------------------------------------------------------------
__global__ void k_node_alpha(const float* __restrict__ h,
                             const float* __restrict__ a_src,
                             const float* __restrict__ a_dst,
                             float* __restrict__ as, float* __restrict__ ad,
                             int N, int H, int C)
{
    int i = blockIdx.x * blockDim.x + threadIdx.x;   // node*H + head
    if (i >= N * H) return;
    int n = i / H, hd = i - n * H;
    const float* hp = h + (long)n * (H * C) + hd * C;
    const float* s0 = a_src + hd * C;
    const float* d0 = a_dst + hd * C;
    float s = 0.f, d = 0.f;
    for (int c = 0; c < C; ++c) { float v = hp[c]; s += v * s0[c]; d += v * d0[c]; }
    as[i] = s;
    ad[i] = d;
}

__global__ void k_fill(float* __restrict__ p, float v, long n)
{
    long i = (long)blockIdx.x * blockDim.x + threadIdx.x;
    if (i < n) p[i] = v;
}

__device__ __forceinline__ void edge_sd(const int* ei, int E, int e, int& s, int& d)
{
    if (e < E) { s = ei[e]; d = ei[E + e]; }
    else       { s = d = e - E; }                 // self loops appended
}

// pass 1: m[d,h] = max over incoming edges of leaky_relu(as[s]+ad[d])
__global__ void k_edge_max(const int* __restrict__ ei, int E, int Etot,
                           const float* __restrict__ as, const float* __restrict__ ad,
                           float* __restrict__ m, int H)
{
    int e = blockIdx.x * blockDim.x + threadIdx.x;
    if (e >= Etot) return;
    int s, d; edge_sd(ei, E, e, s, d);
    for (int hd = 0; hd < H; ++hd) {
        float v = as[s * H + hd] + ad[d * H + hd];
        v = v > 0.f ? v : NEG_SLOPE * v;
        atomicMax(&m[d * H + hd], v);             // global_atomic_max_num_f32
    }
}

// pass 2: den[d,h] = sum exp(e - m[d,h])
__global__ void k_edge_sum(const int* __restrict__ ei, int E, int Etot,
                           const float* __restrict__ as, const float* __restrict__ ad,
                           const float* __restrict__ m, float* __restrict__ den, int H)
{
    int e = blockIdx.x * blockDim.x + threadIdx.x;
    if (e >= Etot) return;
    int s, d; edge_sd(ei, E, e, s, d);
    for (int hd = 0; hd < H; ++hd) {
        float v = as[s * H + hd] + ad[d * H + hd];
        v = v > 0.f ? v : NEG_SLOPE * v;
        atomicAdd(&den[d * H + hd], expf(v - m[d * H + hd]));
    }
}

// pass 3: agg[d,:] += alpha * h[src,:]; one wave per edge, float4 per lane (F=128)
__global__ __launch_bounds__(256) void k_edge_agg(const int* __restrict__ ei,
                                                  int E, int Etot,
                                                  const float* __restrict__ as,
                                                  const float* __restrict__ ad,
                                                  const float* __restrict__ m,
                                                  const float* __restrict__ den,
                                                  const float* __restrict__ h,
                                                  float* __restrict__ agg,
                                                  int H, int C)
{
    int e    = blockIdx.x * 8 + (threadIdx.x >> 5);
    int lane = threadIdx.x & 31;
    if (e >= Etot) return;
    int s, d; edge_sd(ei, E, e, s, d);
    int f  = lane * 4;            // feature group; 4|C so it stays in one head
    int hd = f / C;
    float v = as[s * H + hd] + ad[d * H + hd];
    v = v > 0.f ? v : NEG_SLOPE * v;
    float alpha = expf(v - m[d * H + hd]) / den[d * H + hd];
    const float4 hv = *(const float4*)(h + (long)s * 128 + f);
    float* op = agg + (long)d * 128 + f;
    atomicAdd(op + 0, alpha * hv.x);
    atomicAdd(op + 1, alpha * hv.y);
    atomicAdd(op + 2, alpha * hv.z);
    atomicAdd(op + 3, alpha * hv.w);
}

// layer-1 finalize: out = elu(agg + b1), in place
__global__ void k_final1(float* __restrict__ agg, const float* __restrict__ b, long n)
{
    long i = (long)blockIdx.x * blockDim.x + threadIdx.x;
    if (i >= n) return;
    float v = agg[i] + b[i & 127];
    agg[i] = v > 0.f ? v : expm1f(v);
}

// layer-2 finalize: out += b2 (H==1 so mean over heads is identity)
__global__ void k_final2(float* __restrict__ out, const float* __restrict__ b, long n)
{
    long i = (long)blockIdx.x * blockDim.x + threadIdx.x;
    if (i >= n) return;
    out[i] += b[i & 127];
}

// ---------------------------------------------------------------------------
extern "C" void kernel_launch(void* const* d_in, const int* in_sizes, int n_in,
                              void* d_out, int out_size, void* d_ws, size_t ws_size,
                              hipStream_t stream)
{
    const float* x     = (const float*)d_in[0];
    const int*   ei    = (const int*)  d_in[1];
    const float* W1    = (const float*)d_in[2];
    const float* asrc1 = (const float*)d_in[3];
    const float* adst1 = (const float*)d_in[4];
    const float* b1    = (const float*)d_in[5];
    const float* W2    = (const float*)d_in[6];
    const float* asrc2 = (const float*)d_in[7];
    const float* adst2 = (const float*)d_in[8];
    const float* b2    = (const float*)d_in[9];
    float* out = (float*)d_out;

    const int N    = in_sizes[0] / 128;      // 100000
    const int E    = in_sizes[1] / 2;        // 1600000
    const int Etot = E + N;                  // self loops appended
    const long NF  = (long)N * 128;

    float* ws   = (float*)d_ws;
    float* h    = ws;                        // [N,128]
    float* agg  = ws + NF;                   // [N,128]
    float* as1  = ws + 2 * NF;               // [N,4]
    float* ad1  = as1 + (long)N * 4;
    float* m1   = ad1 + (long)N * 4;
    float* den1 = m1  + (long)N * 4;
    float* as2  = den1 + (long)N * 4;        // [N,1]
    float* ad2  = as2 + N;
    float* m2   = ad2 + N;
    float* den2 = m2  + N;

    const int T = 256;
    dim3 gN16(N / 16);                                   // N % 16 == 0
    dim3 gEdge((Etot + T - 1) / T);
    dim3 gEdgeW((Etot + 7) / 8);
    auto g1 = [&](long n) { return dim3((unsigned)((n + T - 1) / T)); };

    // ---------------- Layer 1 (H=4, C=32) ----------------
    k_gemm_xWT<<<gN16, T, 0, stream>>>(x, W1, h);
    k_node_alpha<<<g1((long)N * 4), T, 0, stream>>>(h, asrc1, adst1, as1, ad1, N, 4, 32);
    k_fill<<<g1((long)N * 4), T, 0, stream>>>(m1, -INFINITY, (long)N * 4);
    k_fill<<<g1((long)N * 4), T, 0, stream>>>(den1, 0.f, (long)N * 4);
    k_fill<<<g1(NF), T, 0, stream>>>(agg, 0.f, NF);
    k_edge_max<<<gEdge, T, 0, stream>>>(ei, E, Etot, as1, ad1, m1, 4);
    k_edge_sum<<<gEdge, T, 0, stream>>>(ei, E, Etot, as1, ad1, m1, den1, 4);
    k_edge_agg<<<gEdgeW, T, 0, stream>>>(ei, E, Etot, as1, ad1, m1, den1, h, agg, 4, 32);
    k_final1<<<g1(NF), T, 0, stream>>>(agg, b1, NF);     // agg now holds layer-2 input

    // ---------------- Layer 2 (H=1, C=128) ----------------
    k_gemm_xWT<<<gN16, T, 0, stream>>>(agg, W2, h);      // reuse h buffer
    k_node_alpha<<<g1(N), T, 0, stream>>>(h, asrc2, adst2, as2, ad2, N, 1, 128);
    k_fill<<<g1(N), T, 0, stream>>>(m2, -INFINITY, N);
    k_fill<<<g1(N), T, 0, stream>>>(den2, 0.f, N);
    k_fill<<<g1(NF), T, 0, stream>>>(out, 0.f, NF);
    k_edge_max<<<gEdge, T, 0, stream>>>(ei, E, Etot, as2, ad2, m2, 1);
    k_edge_sum<<<gEdge, T, 0, stream>>>(ei, E, Etot, as2, ad2, m2, den2, 1);
    k_edge_agg<<<gEdgeW, T, 0, stream>>>(ei, E, Etot, as2, ad2, m2, den2, h, out, 1, 128);
    k_final2<<<g1(NF), T, 0, stream>>>(out, b2, NF);
}